// CGInverterLayer_33414845563318
// MI455X (gfx1250) — compile-verified
//
#include <hip/hip_runtime.h>

// Problem constants (from reference): B is (32, 1, 64, 64) -> (32, 4096), A is 4096x4096.
#define NB   32        // batch rows
#define NN   4096      // vector length / matrix dim
#define NE   (NB*NN)   // 131072 elements per (32,4096) array
#define KS   8         // K-split factor for the GEMM
#define KC   (NN/KS)   // 512 K per slab
#define ITERS 30

typedef float v2f __attribute__((ext_vector_type(2)));
typedef float v8f __attribute__((ext_vector_type(8)));

__device__ __forceinline__ v8f wmma_f32(v2f a, v2f b, v8f c) {
    // V_WMMA_F32_16X16X4_F32: D(16x16,f32) = A(16x4,f32) * B(4x16,f32) + C
    return __builtin_amdgcn_wmma_f32_16x16x4_f32(
        /*neg_a=*/false, a, /*neg_b=*/false, b,
        /*c_mod=*/(short)0, c, /*reuse_a=*/false, /*reuse_b=*/false);
}

// ---------------------------------------------------------------------------
// Y_part[ks] (32 x 4096) = X(32x4096) @ A[kslab](512x4096)
// grid = (NN/256, KS), block = 256 (8 waves). Each wave owns a 32-column
// stripe: two 16-col WMMA tiles sharing the X operands, two 16-row
// accumulators each (M=32). Per k-step: 2x b64 + 4x b32 loads -> 4 WMMAs.
// ---------------------------------------------------------------------------
__global__ __launch_bounds__(256) void cg_gemm(const float* __restrict__ X,
                                               const float* __restrict__ A,
                                               float* __restrict__ Ypart) {
    const int wave = threadIdx.x >> 5;
    const int lane = threadIdx.x & 31;
    const int n    = lane & 15;        // column within tile (B/D lane mapping)
    const int half = lane >> 4;        // 0: K={0,1}/rows M=vgpr, 1: K={2,3}/rows M=vgpr+8
    const int colb = blockIdx.x * 256 + wave * 32;
    const long col = colb + n;

    const int kbeg = blockIdx.y * KC;
    const int kend = kbeg + KC;

    v8f acc00 = {};   // rows 0..15  x cols [colb, colb+16)
    v8f acc10 = {};   // rows 16..31 x cols [colb, colb+16)
    v8f acc01 = {};   // rows 0..15  x cols [colb+16, colb+32)
    v8f acc11 = {};   // rows 16..31 x cols [colb+16, colb+32)

    #pragma unroll 4
    for (int k = kbeg; k < kend; k += 4) {
        const int kk = k + 2 * half;
        // A-operand (X tile): contiguous float2 along K -> global_load_b64
        v2f a0 = *(const v2f*)(X + (long)n * NN + kk);
        v2f a1 = *(const v2f*)(X + (long)(n + 16) * NN + kk);
        // B-operands (two 16-col slabs of A), rows kk and kk+1
        const float* arow0 = A + (long)kk * NN + col;
        const float* arow1 = arow0 + NN;
        v2f b0, b1;
        b0.x = arow0[0];  b0.y = arow1[0];
        b1.x = arow0[16]; b1.y = arow1[16];
        acc00 = wmma_f32(a0, b0, acc00);
        acc10 = wmma_f32(a1, b0, acc10);
        acc01 = wmma_f32(a0, b1, acc01);
        acc11 = wmma_f32(a1, b1, acc11);
    }

    float* Y = Ypart + (long)blockIdx.y * NE;
    #pragma unroll
    for (int i = 0; i < 8; ++i) {
        const int row = i + 8 * half;                 // D layout: M = vgpr + 8*(lane/16)
        Y[(long)row * NN + col]             = acc00[i];
        Y[(long)(row + 16) * NN + col]      = acc10[i];
        Y[(long)row * NN + col + 16]        = acc01[i];
        Y[(long)(row + 16) * NN + col + 16] = acc11[i];
    }
}

// AP[idx] = sum over the KS partials (fixed order -> deterministic)
__global__ __launch_bounds__(256) void cg_reduce_parts(const float* __restrict__ part,
                                                       float* __restrict__ AP) {
    const int idx = blockIdx.x * 256 + threadIdx.x;
    float s = 0.f;
    #pragma unroll
    for (int j = 0; j < KS; ++j) s += part[(long)j * NE + idx];
    AP[idx] = s;
}

// X = B; R = B - (B@A); P = R
__global__ __launch_bounds__(256) void cg_init(const float* __restrict__ B,
                                               const float* __restrict__ AB,
                                               float* __restrict__ X,
                                               float* __restrict__ R,
                                               float* __restrict__ P) {
    const int idx = blockIdx.x * 256 + threadIdx.x;
    const float b = B[idx];
    const float r = b - AB[idx];
    X[idx] = b; R[idx] = r; P[idx] = r;
}

// denom[b] = dot(P[b], AP[b]); alpha[b] = rz[b] / where(denom==0, 1e-8, denom)
__global__ __launch_bounds__(256) void cg_dot_alpha(const float* __restrict__ P,
                                                    const float* __restrict__ AP,
                                                    const float* __restrict__ rz,
                                                    float* __restrict__ alpha) {
    __shared__ float sm[256];
    const int b = blockIdx.x, t = threadIdx.x;
    const float* p  = P  + (long)b * NN;
    const float* ap = AP + (long)b * NN;
    float s = 0.f;
    for (int i = t; i < NN; i += 256) s += p[i] * ap[i];
    sm[t] = s; __syncthreads();
    for (int o = 128; o > 0; o >>= 1) { if (t < o) sm[t] += sm[t + o]; __syncthreads(); }
    if (t == 0) {
        float d = sm[0];
        if (d == 0.f) d = 1e-8f;
        alpha[b] = rz[b] / d;
    }
}

// X += alpha*P ; R -= alpha*AP
__global__ __launch_bounds__(256) void cg_update_xr(float* __restrict__ X,
                                                    float* __restrict__ R,
                                                    const float* __restrict__ P,
                                                    const float* __restrict__ AP,
                                                    const float* __restrict__ alpha) {
    const int idx = blockIdx.x * 256 + threadIdx.x;
    const float a = alpha[idx >> 12];
    X[idx] = fmaf(a, P[idx], X[idx]);
    R[idx] = fmaf(-a, AP[idx], R[idx]);
}

// s = dot(R,R). first: rz = s. else: beta = s / where(rz==0,1e-8,rz); rz = s.
__global__ __launch_bounds__(256) void cg_rz(const float* __restrict__ R,
                                             float* __restrict__ rz,
                                             float* __restrict__ beta,
                                             int first) {
    __shared__ float sm[256];
    const int b = blockIdx.x, t = threadIdx.x;
    const float* r = R + (long)b * NN;
    float s = 0.f;
    for (int i = t; i < NN; i += 256) { const float v = r[i]; s += v * v; }
    sm[t] = s; __syncthreads();
    for (int o = 128; o > 0; o >>= 1) { if (t < o) sm[t] += sm[t + o]; __syncthreads(); }
    if (t == 0) {
        const float sn = sm[0];
        if (first) {
            rz[b] = sn;
        } else {
            float d = rz[b];
            if (d == 0.f) d = 1e-8f;
            beta[b] = sn / d;
            rz[b] = sn;
        }
    }
}

// P = R + beta*P
__global__ __launch_bounds__(256) void cg_update_p(float* __restrict__ P,
                                                   const float* __restrict__ R,
                                                   const float* __restrict__ beta) {
    const int idx = blockIdx.x * 256 + threadIdx.x;
    P[idx] = fmaf(beta[idx >> 12], P[idx], R[idx]);
}

extern "C" void kernel_launch(void* const* d_in, const int* in_sizes, int n_in,
                              void* d_out, int out_size, void* d_ws, size_t ws_size,
                              hipStream_t stream) {
    const float* B = (const float*)d_in[0];   // (32, 4096) flattened
    const float* A = (const float*)d_in[1];   // (4096, 4096)
    float* X = (float*)d_out;                 // solution, (32, 4096)

    float* ws    = (float*)d_ws;
    float* R     = ws;                        // NE floats
    float* P     = ws + (long)NE;             // NE floats
    float* AP    = ws + 2L * NE;              // NE floats
    float* part  = ws + 3L * NE;              // KS*NE floats (GEMM K-split partials)
    float* rz    = part + (long)KS * NE;      // NB floats
    float* alpha = rz + NB;                   // NB floats
    float* beta  = alpha + NB;                // NB floats
    (void)ws_size; (void)n_in; (void)in_sizes; (void)out_size;

    const dim3 gGemm(NN / 256, KS), bGemm(256);
    const dim3 gElem(NE / 256), bElem(256);
    const dim3 gRow(NB), bRow(256);

    // --- init: AP = B @ A ; X = B ; R = B - AP ; P = R ; rz = sum(R*R) ---
    cg_gemm<<<gGemm, bGemm, 0, stream>>>(B, A, part);
    cg_reduce_parts<<<gElem, bElem, 0, stream>>>(part, AP);
    cg_init<<<gElem, bElem, 0, stream>>>(B, AP, X, R, P);
    cg_rz<<<gRow, bRow, 0, stream>>>(R, rz, beta, /*first=*/1);

    // --- 30 fixed CG iterations ---
    for (int it = 0; it < ITERS; ++it) {
        cg_gemm<<<gGemm, bGemm, 0, stream>>>(P, A, part);
        cg_reduce_parts<<<gElem, bElem, 0, stream>>>(part, AP);
        cg_dot_alpha<<<gRow, bRow, 0, stream>>>(P, AP, rz, alpha);
        cg_update_xr<<<gElem, bElem, 0, stream>>>(X, R, P, AP, alpha);
        cg_rz<<<gRow, bRow, 0, stream>>>(R, rz, beta, /*first=*/0);
        cg_update_p<<<gElem, bElem, 0, stream>>>(P, R, beta);
    }
}